// ResGCN_28509992911040
// MI455X (gfx1250) — compile-verified
//
#include <hip/hip_runtime.h>

#define N_NODES 50000
#define N_EDGES 1600000
#define D_IN 128
#define HID 64

typedef __attribute__((ext_vector_type(2))) float v2f;
typedef __attribute__((ext_vector_type(8))) float v8f;

// ---------------- degree / normalization ----------------

__global__ void k_init_deg(float* __restrict__ deg, int n) {
    int i = blockIdx.x * blockDim.x + threadIdx.x;
    if (i < n) deg[i] = 1.0f;  // self-loop edge weight
}

__global__ void k_deg_accum(const int* __restrict__ dst, const float* __restrict__ ew,
                            float* __restrict__ deg, int e) {
    int i = blockIdx.x * blockDim.x + threadIdx.x;
    if (i < e) unsafeAtomicAdd(&deg[dst[i]], ew[i]);
}

__global__ void k_deg_to_dinv(float* __restrict__ deg, int n) {
    int i = blockIdx.x * blockDim.x + threadIdx.x;
    if (i < n) {
        float d = deg[i];
        deg[i] = (d > 0.0f) ? rsqrtf(d) : 0.0f;
    }
}

// ---------------- dense linear: H = A @ W.T via V_WMMA_F32_16X16X4_F32 ----------------
// A: [M,K] row-major, W: [HID,K] row-major, H: [M,HID]. One wave -> 16x64 tile.

template <int K>
__launch_bounds__(256)
__global__ void k_gemm_wmma(const float* __restrict__ A, const float* __restrict__ W,
                            float* __restrict__ H, int M) {
    __shared__ float sW[HID * K];
    for (int i = threadIdx.x; i < HID * K; i += 256) sW[i] = W[i];
    __syncthreads();

    const int wave = threadIdx.x >> 5;
    const int lane = threadIdx.x & 31;
    const int half = lane >> 4;   // 0: K pair {k,k+1}, 1: K pair {k+2,k+3}
    const int r    = lane & 15;   // row (A) / col (B) within tile
    const int row0 = (blockIdx.x * 8 + wave) * 16;
    if (row0 >= M) return;

    v8f acc0 = {}, acc1 = {}, acc2 = {}, acc3 = {};
    const float* arow = A + (size_t)(row0 + r) * K;

#pragma unroll 4
    for (int k = 0; k < K; k += 4) {
        const int ko = k + 2 * half;
        v2f a  = *(const v2f*)(arow + ko);
        v2f b0 = *(const v2f*)(&sW[( 0 + r) * K + ko]);
        v2f b1 = *(const v2f*)(&sW[(16 + r) * K + ko]);
        v2f b2 = *(const v2f*)(&sW[(32 + r) * K + ko]);
        v2f b3 = *(const v2f*)(&sW[(48 + r) * K + ko]);
        acc0 = __builtin_amdgcn_wmma_f32_16x16x4_f32(false, a, false, b0, (short)0, acc0, false, false);
        acc1 = __builtin_amdgcn_wmma_f32_16x16x4_f32(false, a, false, b1, (short)0, acc1, false, false);
        acc2 = __builtin_amdgcn_wmma_f32_16x16x4_f32(false, a, false, b2, (short)0, acc2, false, false);
        acc3 = __builtin_amdgcn_wmma_f32_16x16x4_f32(false, a, false, b3, (short)0, acc3, false, false);
    }

    // D layout: VGPR v, lanes 0-15 -> M=v ; lanes 16-31 -> M=v+8 ; N = lane&15
#pragma unroll
    for (int v = 0; v < 8; ++v) {
        float* p = H + (size_t)(row0 + v + 8 * half) * HID;
        p[ 0 + r] = acc0[v];
        p[16 + r] = acc1[v];
        p[32 + r] = acc2[v];
        p[48 + r] = acc3[v];
    }
}

// ---------------- aggregation (gather-scale-scatter) ----------------

// agg[n,:] = h[n,:] * dinv[n]^2   (self-loop contribution, ew=1)
__global__ void k_agg_self(const float* __restrict__ h, const float* __restrict__ dinv,
                           float* __restrict__ agg, int total) {
    int i = blockIdx.x * blockDim.x + threadIdx.x;
    if (i < total) {
        float s = dinv[i >> 6];
        agg[i] = h[i] * s * s;
    }
}

// one wave per edge; each lane moves 2 floats of the 64-wide message
__global__ void k_agg_edges(const int* __restrict__ src, const int* __restrict__ dst,
                            const float* __restrict__ ew, const float* __restrict__ dinv,
                            const float* __restrict__ h, float* __restrict__ agg, int e) {
    int t = blockIdx.x * blockDim.x + threadIdx.x;
    int eid = t >> 5;
    if (eid >= e) return;
    int lane = t & 31;
    int s = src[eid], d = dst[eid];
    float norm = dinv[s] * ew[eid] * dinv[d];
    v2f hv = *(const v2f*)(h + (size_t)s * HID + lane * 2);
    float* p = agg + (size_t)d * HID + lane * 2;
    unsafeAtomicAdd(p,     hv.x * norm);
    unsafeAtomicAdd(p + 1, hv.y * norm);
}

__global__ void k_relu_bias(float* __restrict__ a, const float* __restrict__ b, int total) {
    int i = blockIdx.x * blockDim.x + threadIdx.x;
    if (i < total) {
        float v = a[i] + b[i & (HID - 1)];
        a[i] = v > 0.0f ? v : 0.0f;
    }
}

__global__ void k_bias_residual(float* __restrict__ out, const float* __restrict__ b,
                                const float* __restrict__ h1, int total) {
    int i = blockIdx.x * blockDim.x + threadIdx.x;
    if (i < total) out[i] = out[i] + b[i & (HID - 1)] + h1[i];
}

// ---------------- driver ----------------

extern "C" void kernel_launch(void* const* d_in, const int* in_sizes, int n_in,
                              void* d_out, int out_size, void* d_ws, size_t ws_size,
                              hipStream_t stream) {
    const float* x  = (const float*)d_in[0];
    const int*   ei = (const int*)  d_in[1];
    const float* ew = (const float*)d_in[2];
    const float* W1 = (const float*)d_in[3];
    const float* b1 = (const float*)d_in[4];
    const float* W2 = (const float*)d_in[5];
    const float* b2 = (const float*)d_in[6];
    float* out = (float*)d_out;

    const int n = N_NODES, e = N_EDGES;
    const int* src = ei;          // ei[0,:]
    const int* dst = ei + e;      // ei[1,:]

    float* dinv = (float*)d_ws;                    // [n]
    float* hbuf = dinv + n;                        // [n*HID]
    float* h1   = hbuf + (size_t)n * HID;          // [n*HID]

    const int total = n * HID;
    const int elemBlocks = (total + 255) / 256;
    const int edgeBlocks = (int)(((size_t)e * 32 + 255) / 256);
    const int rowTiles = (n + 15) / 16;
    const int gemmBlocks = (rowTiles + 7) / 8;     // 8 waves (16-row tiles) per block

    // normalization: deg (with self loop) -> deg^{-1/2}
    k_init_deg   <<<(n + 255) / 256, 256, 0, stream>>>(dinv, n);
    k_deg_accum  <<<(e + 255) / 256, 256, 0, stream>>>(dst, ew, dinv, e);
    k_deg_to_dinv<<<(n + 255) / 256, 256, 0, stream>>>(dinv, n);

    // layer 1: h = x @ W1.T ; h1 = relu(scatter(norm * gather(h)) + b1)
    k_gemm_wmma<D_IN><<<gemmBlocks, 256, 0, stream>>>(x, W1, hbuf, n);
    k_agg_self <<<elemBlocks, 256, 0, stream>>>(hbuf, dinv, h1, total);
    k_agg_edges<<<edgeBlocks, 256, 0, stream>>>(src, dst, ew, dinv, hbuf, h1, e);
    k_relu_bias<<<elemBlocks, 256, 0, stream>>>(h1, b1, total);

    // layer 2: h = h1 @ W2.T ; out = scatter(norm * gather(h)) + b2 + h1
    k_gemm_wmma<HID><<<gemmBlocks, 256, 0, stream>>>(h1, W2, hbuf, n);
    k_agg_self <<<elemBlocks, 256, 0, stream>>>(hbuf, dinv, out, total);
    k_agg_edges<<<edgeBlocks, 256, 0, stream>>>(src, dst, ew, dinv, hbuf, out, e);
    k_bias_residual<<<elemBlocks, 256, 0, stream>>>(out, b2, h1, total);
}